// DechirpSTFT_70059506532907
// MI455X (gfx1250) — compile-verified
//
#include <hip/hip_runtime.h>
#include <math.h>

// DechirpSTFT on gfx1250: four-step radix-32 FFT (1024 = 32x32) where each
// DFT-32 stage is a 32x32x32 real GEMM executed with V_WMMA_F32_16X16X4_F32.
// Output (268 MB, write-once) is streamed with non-temporal stores.

typedef __attribute__((ext_vector_type(2))) float v2f;
typedef __attribute__((ext_vector_type(8))) float v8f;

#define KFFT 1024
#define HOPSZ 512
#define RDX 32          // radix
#define PD 33           // padded LDS stride (kills bank conflicts)
#define WAVES 8
#define TPB 256

__device__ __forceinline__ v2f fragA(const float* A, int ld, int mo, int ko,
                                     int lrow, int kb) {
    // 16x4 fp32 A tile: lanes 0-15 hold K={ko+0,ko+1}, lanes 16-31 K={ko+2,ko+3}
    v2f a;
    a.x = A[(mo + lrow) * ld + ko + kb];
    a.y = A[(mo + lrow) * ld + ko + kb + 1];
    return a;
}

__device__ __forceinline__ v2f fragB(const float* B, int ld, int ko, int no,
                                     int lrow, int kb) {
    // 4x16 fp32 B tile: lanes 0-15 hold rows {ko+0,ko+1}, lanes 16-31 rows {ko+2,ko+3}
    v2f b;
    b.x = B[(ko + kb) * ld + no + lrow];
    b.y = B[(ko + kb + 1) * ld + no + lrow];
    return b;
}

__device__ __forceinline__ v8f wmma_f32(v2f a, v2f b, v8f c) {
    // (neg_a, A, neg_b, B, c_mod, C, reuse_a, reuse_b)
    return __builtin_amdgcn_wmma_f32_16x16x4_f32(false, a, false, b,
                                                 (short)0, c, false, false);
}

__global__ __launch_bounds__(TPB)
void DechirpSTFT_70059506532907_kernel(const float* __restrict__ x,
                                       const float* __restrict__ dlnf,
                                       float* __restrict__ out, int nw) {
    extern __shared__ float smem[];
    float* wd   = smem;                 // 1024 : Hann-windowed samples
    float* C32  = wd + KFFT;            // 32x33: cos(2*pi*a*b/32)
    float* S32  = C32 + RDX * PD;       // 32x33: sin(2*pi*a*b/32)
    float* TWc  = S32 + RDX * PD;       // 32x32: cos twiddles W1024
    float* TWs  = TWc + RDX * RDX;      // 32x32: sin twiddles W1024
    float* wbase = TWs + RDX * RDX;     // per-wave scratch: 3 * 32x33 each

    const int tid  = threadIdx.x;
    const int w    = blockIdx.x;
    const int lane = tid & 31;
    const int wv   = tid >> 5;
    const int lrow = lane & 15;
    const int lhi  = lane >> 4;
    const int kb   = lhi * 2;   // A/B fragment K sub-offset
    const int mh   = lhi * 8;   // C/D fragment M sub-offset

    const float c2pi = 6.28318530717958647692f;

    // ---- shared setup: windowed input, DFT-32 matrix, twiddle table ----
    const float* xw = x + (size_t)w * HOPSZ;
    for (int j = tid; j < KFFT; j += TPB) {
        float h = 0.5f * (1.0f - __cosf(c2pi * (float)j * (1.0f / (float)KFFT)));
        wd[j] = xw[j] * h;
    }
    for (int e = tid; e < RDX * RDX; e += TPB) {
        int a = e >> 5, b = e & 31;
        float th = c2pi * (float)(a * b) * (1.0f / (float)RDX);
        float sn, cs;
        __sincosf(th, &sn, &cs);
        C32[a * PD + b] = cs;
        S32[a * PD + b] = sn;
        float th2 = c2pi * (float)(a * b) * (1.0f / (float)KFFT);
        __sincosf(th2, &sn, &cs);
        TWc[e] = cs;
        TWs[e] = sn;
    }
    __syncthreads();

    float* X  = wbase + wv * (3 * RDX * PD);  // resampled real input, X[n1][n2]
    float* Yr = X + RDX * PD;                 // stage-1 output (real)
    float* Yi = Yr + RDX * PD;                // stage-1 output (imag)

    for (int bi = 0; bi < 2; ++bi) {
        const int b = wv + bi * WAVES;        // beta index (16 total, 2 per wave)

        // ---- time-warp resample (matches reference math, fp32) ----
        float beta = 2.0f * dlnf[b];
        bool  safe = fabsf(beta) < 1e-8f;
        float bs   = safe ? 1e-8f : beta;
        float em1  = expf(bs) - 1.0f;
        float sc   = 2.0f / bs;
        #pragma unroll 4
        for (int t = 0; t < RDX; ++t) {
            int j = lane + RDX * t;           // n = n1 + 32*n2; n1=lane, n2=t
            float tau  = (float)j * (1.0f / 1023.0f);
            float tsrc = safe ? (2.0f * tau - 1.0f)
                              : (sc * log1pf(tau * em1) - 1.0f);
            float pos = (tsrc + 1.0f) * 0.5f * 1023.0f;
            int lo = (int)pos;
            lo = lo < 0 ? 0 : (lo > KFFT - 2 ? KFFT - 2 : lo);
            float fr = pos - (float)lo;
            X[lane * PD + t] = wd[lo] * (1.0f - fr) + wd[lo + 1] * fr;
        }

        // ---- stage 1: Y = X * (C - iS); X real -> Yr = X*C, Yi = -(X*S) ----
        for (int ti = 0; ti < 2; ++ti)
        for (int tj = 0; tj < 2; ++tj) {
            v8f accr = {};
            v8f accp = {};
            #pragma unroll
            for (int kk = 0; kk < 8; ++kk) {
                int ko = kk * 4;
                v2f a   = fragA(X,   PD, ti * 16, ko, lrow, kb);
                v2f bc  = fragB(C32, PD, ko, tj * 16, lrow, kb);
                v2f bsn = fragB(S32, PD, ko, tj * 16, lrow, kb);
                accr = wmma_f32(a, bc, accr);
                accp = wmma_f32(a, bsn, accp);
            }
            #pragma unroll
            for (int v = 0; v < 8; ++v) {
                int row = ti * 16 + mh + v;
                int col = tj * 16 + lrow;
                Yr[row * PD + col] = accr[v];
                Yi[row * PD + col] = -accp[v];
            }
        }

        // ---- twiddle: Y' = Y * (TWc - i*TWs), elementwise ----
        #pragma unroll 4
        for (int t = 0; t < RDX; ++t) {
            int idx = t * PD + lane;          // n1 = t, k2 = lane
            float tc = TWc[t * RDX + lane];
            float ts = TWs[t * RDX + lane];
            float yr = Yr[idx], yi = Yi[idx];
            Yr[idx] = yr * tc + yi * ts;
            Yi[idx] = yi * tc - yr * ts;
        }

        // ---- stage 2: out = (C - iS) * Y'; write interleaved complex64 ----
        size_t obase = ((size_t)b * (size_t)nw + (size_t)w) * (size_t)KFFT;
        for (int ti = 0; ti < 2; ++ti)
        for (int tj = 0; tj < 2; ++tj) {
            v8f accr = {};
            v8f acci = {};
            #pragma unroll
            for (int kk = 0; kk < 8; ++kk) {
                int ko = kk * 4;
                v2f ac  = fragA(C32, PD, ti * 16, ko, lrow, kb);
                v2f as  = fragA(S32, PD, ti * 16, ko, lrow, kb);
                v2f br  = fragB(Yr, PD, ko, tj * 16, lrow, kb);
                v2f bim = fragB(Yi, PD, ko, tj * 16, lrow, kb);
                accr = wmma_f32(ac, br, accr);
                accr = wmma_f32(as, bim, accr);
                acci = wmma_f32(ac, bim, acci);
                v2f nas = -as;                // f32 WMMA has no A-neg modifier
                acci = wmma_f32(nas, br, acci);
            }
            #pragma unroll
            for (int v = 0; v < 8; ++v) {
                int k1 = ti * 16 + mh + v;
                int k2 = tj * 16 + lrow;
                size_t o = obase + (size_t)(k1 * RDX + k2);
                v2f val;                      // native ext-vector (not HIP float2)
                val.x = accr[v];
                val.y = acci[v];
                // write-once 268 MB stream: non-temporal (global_store_b64 th:NT)
                __builtin_nontemporal_store(val,
                    reinterpret_cast<v2f*>(out + 2 * o));
            }
        }
    }
}

extern "C" void kernel_launch(void* const* d_in, const int* in_sizes, int n_in,
                              void* d_out, int out_size, void* d_ws, size_t ws_size,
                              hipStream_t stream) {
    const float* x    = (const float*)d_in[0];
    const float* dlnf = (const float*)d_in[1];
    float* out = (float*)d_out;

    int N  = in_sizes[0];
    int nw = (N - KFFT) / HOPSZ + 1;   // 2047 for N = 1048576

    size_t shmem = (size_t)(KFFT            // wd
                          + 2 * RDX * PD    // C32, S32
                          + 2 * RDX * RDX   // TWc, TWs
                          + WAVES * 3 * RDX * PD) * sizeof(float); // ~119 KB
    DechirpSTFT_70059506532907_kernel<<<nw, TPB, shmem, stream>>>(x, dlnf, out, nw);
}